// GraphConvLayer_55783035240593
// MI455X (gfx1250) — compile-verified
//
#include <hip/hip_runtime.h>

typedef __attribute__((ext_vector_type(2))) float v2f;
typedef __attribute__((ext_vector_type(8))) float v8f;

#define D_DIM 128
#define MT 4  // 16-row M-tiles per block

// ---------------------------------------------------------------- zero agg
__global__ __launch_bounds__(256) void gc_zero_kernel(float4* __restrict__ p, int n4) {
    int i = blockIdx.x * blockDim.x + threadIdx.x;
    if (i < n4) p[i] = make_float4(0.f, 0.f, 0.f, 0.f);
}

// Guaranteed-native no-return f32 atomic add at device scope (L2 atomic units).
__device__ __forceinline__ void atomic_add_f32_noret(float* p, float v) {
    asm volatile("global_atomic_add_f32 %0, %1, off scope:SCOPE_DEV"
                 :: "v"(p), "v"(v) : "memory");
}

// ------------------------------------------------- edge scatter: agg[dst] += x[src]
// one wave32 per edge; lane owns 4 consecutive columns (float4 coalesced load,
// 4 native no-return global_atomic_add_f32)
__global__ __launch_bounds__(256) void gc_scatter_kernel(
    const float* __restrict__ x,
    const long long* __restrict__ src,
    const long long* __restrict__ dst,
    float* __restrict__ agg,
    int n_edges)
{
    const int lane   = threadIdx.x & 31;
    const int waveId = blockIdx.x * (blockDim.x >> 5) + (threadIdx.x >> 5);
    if (waveId >= n_edges) return;

    const long long s = src[waveId];
    const long long d = dst[waveId];

    const float4 v = *(const float4*)(x + s * D_DIM + lane * 4);
    float* p = agg + d * D_DIM + lane * 4;
    atomic_add_f32_noret(p + 0, v.x);
    atomic_add_f32_noret(p + 1, v.y);
    atomic_add_f32_noret(p + 2, v.z);
    atomic_add_f32_noret(p + 3, v.w);
}

// ------------------------------------------------- fused dual GEMM + bias + ReLU
// OUT = relu(AGG @ Wrel^T + X @ Wroot^T + b), via V_WMMA_F32_16X16X4_F32.
// Block = 256 threads = 8 waves; wave w handles output columns n0 = 16*w.
// Block covers MT*16 = 64 output rows; accumulators kept across both GEMMs.
__global__ __launch_bounds__(256) void gc_gemm_kernel(
    const float* __restrict__ agg,
    const float* __restrict__ x,
    const float* __restrict__ Wrel,
    const float* __restrict__ Wroot,
    const float* __restrict__ brel,
    float* __restrict__ out,
    int n_nodes)
{
    const int lane  = threadIdx.x & 31;
    const int wave  = threadIdx.x >> 5;     // 0..7 -> N tile
    const int l     = lane & 15;
    const int half  = lane >> 4;
    const int n0    = wave * 16;
    const int mbase = blockIdx.x * (16 * MT);
    const int koff  = half * 2;             // K = 2*half + vgpr for A and B frags

    v8f acc[MT] = {};                        // f32 accumulators (C/D 16x16)

    // clamped A-row per tile: keeps all lanes loading valid memory so EXEC
    // stays all-1s through the WMMA loop (tail rows predicated at store)
    int arow[MT];
#pragma unroll
    for (int t = 0; t < MT; ++t) {
        int m = mbase + t * 16 + l;
        arow[t] = (m < n_nodes) ? m : (n_nodes - 1);
    }
    const int wrow = n0 + l;                 // B = W^T -> read row n of W

    for (int k = 0; k < D_DIM; k += 4) {
        const v2f b_rel  = *(const v2f*)(Wrel  + wrow * D_DIM + k + koff);
        const v2f b_root = *(const v2f*)(Wroot + wrow * D_DIM + k + koff);
#pragma unroll
        for (int t = 0; t < MT; ++t) {
            const v2f a_agg = *(const v2f*)(agg + arow[t] * D_DIM + k + koff);
            const v2f a_x   = *(const v2f*)(x   + arow[t] * D_DIM + k + koff);
            acc[t] = __builtin_amdgcn_wmma_f32_16x16x4_f32(
                false, a_agg, false, b_rel, (short)0, acc[t], false, false);
            acc[t] = __builtin_amdgcn_wmma_f32_16x16x4_f32(
                false, a_x, false, b_root, (short)0, acc[t], false, false);
        }
    }

    const float bias = brel[n0 + l];
#pragma unroll
    for (int t = 0; t < MT; ++t) {
        const int mrow0 = mbase + t * 16 + half * 8;  // C/D: M = 8*half + r
#pragma unroll
        for (int r = 0; r < 8; ++r) {
            const int m = mrow0 + r;
            if (m < n_nodes) {
                float v = acc[t][r] + bias;
                out[m * D_DIM + n0 + l] = v > 0.f ? v : 0.f;
            }
        }
    }
}

extern "C" void kernel_launch(void* const* d_in, const int* in_sizes, int n_in,
                              void* d_out, int out_size, void* d_ws, size_t ws_size,
                              hipStream_t stream) {
    const float*     x     = (const float*)d_in[0];
    const long long* eidx  = (const long long*)d_in[1];   // int64 [2, E]
    const float*     Wrel  = (const float*)d_in[2];
    const float*     brel  = (const float*)d_in[3];
    const float*     Wroot = (const float*)d_in[4];
    float*           out   = (float*)d_out;

    const int n_nodes = in_sizes[0] / D_DIM;
    const int n_edges = in_sizes[1] / 2;

    const long long* src = eidx;             // edge_index[0]
    const long long* dst = eidx + n_edges;   // edge_index[1]

    float* agg = (float*)d_ws;               // [n_nodes, 128] f32 scratch

    // 1) zero aggregation buffer (workspace is poisoned by harness)
    {
        const int n4 = n_nodes * (D_DIM / 4);
        gc_zero_kernel<<<(n4 + 255) / 256, 256, 0, stream>>>((float4*)agg, n4);
    }

    // 2) scatter-add: one wave per edge (8 waves / 256-thread block)
    {
        const int wavesPerBlock = 256 / 32;
        const int blocks = (n_edges + wavesPerBlock - 1) / wavesPerBlock;
        gc_scatter_kernel<<<blocks, 256, 0, stream>>>(x, src, dst, agg, n_edges);
    }

    // 3) fused dual WMMA GEMM + bias + ReLU
    {
        const int rowsPerBlock = 16 * MT;
        const int blocks = (n_nodes + rowsPerBlock - 1) / rowsPerBlock;
        gc_gemm_kernel<<<blocks, 256, 0, stream>>>(agg, x, Wrel, Wroot, brel, out, n_nodes);
    }
}